// FusedBlock0_7_68109591380139
// MI455X (gfx1250) — compile-verified
//
#include <hip/hip_runtime.h>

typedef __attribute__((ext_vector_type(2))) float v2f;
typedef __attribute__((ext_vector_type(8))) float v8f;

#define IN_CH 128
#define OUT_CH 256
#define APITCH 260   // 256 + 4 pad: A rows land on distinct LDS banks
#define MTILE 64     // nodes per block: 4x weight reuse vs 16

// ---------------- Kernel 1: zero workspace ----------------
__global__ __launch_bounds__(256) void zero_ws_kernel(float* __restrict__ ws, long long n) {
  long long i = (long long)blockIdx.x * blockDim.x + threadIdx.x;
  long long stride = (long long)gridDim.x * blockDim.x;
  for (; i < n; i += stride) ws[i] = 0.0f;
}

// ---------------- Kernel 2: edge scatter-add (one wave32 per edge) ----------------
__global__ __launch_bounds__(256) void scatter_kernel(const float* __restrict__ x,
                                                      const long long* __restrict__ ei,
                                                      float* __restrict__ sum_agg,
                                                      float* __restrict__ count,
                                                      int num_edges) {
  int gid  = blockIdx.x * blockDim.x + threadIdx.x;
  int e    = gid >> 5;
  int lane = gid & 31;
  if (e >= num_edges) return;
  long long src = ei[e];
  long long dst = ei[num_edges + e];
  // each lane owns 4 contiguous channels -> coalesced b128 gather; atomics land in L2
  const float4 v = *(const float4*)(x + src * IN_CH + lane * 4);
  float* base = sum_agg + dst * IN_CH + lane * 4;
  atomicAdd(base + 0, v.x);
  atomicAdd(base + 1, v.y);
  atomicAdd(base + 2, v.z);
  atomicAdd(base + 3, v.w);
  if (lane == 0) atomicAdd(count + dst, 1.0f);
}

// ---------------- Kernel 3: fused mean + dual GEMM + bias + relu ----------------
// Block: 64 nodes (M), all 256 out cols. 8 waves; each wave owns 32 cols (2 N-subtiles)
// and sweeps 4 M-subtiles, so each B fragment is reused by 4 WMMAs.
// Combined contraction: out = [mean ; x] (K=256) @ [W_l ; W_r]^T
__global__ __launch_bounds__(256) void sage_gemm_kernel(const float* __restrict__ x,
                                                        const float* __restrict__ sum_agg,
                                                        const float* __restrict__ count,
                                                        const float* __restrict__ W_l,
                                                        const float* __restrict__ b_l,
                                                        const float* __restrict__ W_r,
                                                        float* __restrict__ out,
                                                        int num_nodes) {
  __shared__ float Atile[MTILE * APITCH];
  __shared__ float inv_cnt[MTILE];

  const int tid = threadIdx.x;
  const int node_base = blockIdx.x * MTILE;

  if (tid < MTILE) {
    int n = node_base + tid;
    if (n >= num_nodes) n = num_nodes - 1;
    inv_cnt[tid] = 1.0f / fmaxf(count[n], 1.0f);
  }
  __syncthreads();

  // Cooperative fill: thread t owns column t for all 64 rows (coalesced in global).
  // cols 0..127 = mean (sum * 1/count), cols 128..255 = x
  {
    const int col = tid;
#pragma unroll 8
    for (int row = 0; row < MTILE; ++row) {
      int n = node_base + row;
      if (n >= num_nodes) n = num_nodes - 1;
      float v;
      if (col < IN_CH) v = sum_agg[(long long)n * IN_CH + col] * inv_cnt[row];
      else             v = x[(long long)n * IN_CH + (col - IN_CH)];
      Atile[row * APITCH + col] = v;
    }
  }
  __syncthreads();

  const int wid   = tid >> 5;
  const int lane  = tid & 31;
  const int rc    = lane & 15;             // row (A) / col (B,C,D) within a 16-tile
  const int khalf = (lane < 16) ? 0 : 2;   // 16x4 fp32 layout: hi lanes hold K+2..K+3
  const int cb0 = wid * 32;                // this wave's two 16-col output subtiles
  const int cb1 = cb0 + 16;

  v8f acc[4][2];
#pragma unroll
  for (int mt = 0; mt < 4; ++mt) { acc[mt][0] = (v8f){}; acc[mt][1] = (v8f){}; }

#pragma unroll
  for (int half = 0; half < 2; ++half) {
    const float* __restrict__ W = half ? W_r : W_l;          // [256 out][128 in] row-major
    const float* __restrict__ w0 = W + (long long)(cb0 + rc) * IN_CH;
    const float* __restrict__ w1 = W + (long long)(cb1 + rc) * IN_CH;
    const float* __restrict__ a  = Atile + rc * APITCH + half * IN_CH;
#pragma unroll 2
    for (int k0 = 0; k0 < IN_CH; k0 += 4) {
      const int kk = k0 + khalf;
      v2f b0 = *(const v2f*)(w0 + kk);     // global_load_b64 (L2 resident), reused 4x
      v2f b1 = *(const v2f*)(w1 + kk);
#pragma unroll
      for (int mt = 0; mt < 4; ++mt) {
        v2f av = *(const v2f*)(a + mt * (16 * APITCH) + kk);  // ds_load_b64
        acc[mt][0] = __builtin_amdgcn_wmma_f32_16x16x4_f32(false, av, false, b0,
                                                           (short)0, acc[mt][0], false, false);
        acc[mt][1] = __builtin_amdgcn_wmma_f32_16x16x4_f32(false, av, false, b1,
                                                           (short)0, acc[mt][1], false, false);
      }
    }
  }

  // Epilogue: bias + relu + store per the 16x16 f32 C/D layout
  // (VGPR r: lanes 0-15 -> M=r, lanes 16-31 -> M=r+8; N = lane%16)
  const float bias0 = b_l[cb0 + rc];
  const float bias1 = b_l[cb1 + rc];
  const int mofs = (lane < 16) ? 0 : 8;
#pragma unroll
  for (int mt = 0; mt < 4; ++mt) {
#pragma unroll
    for (int r = 0; r < 8; ++r) {
      const int m = node_base + mt * 16 + r + mofs;
      if (m < num_nodes) {
        out[(long long)m * OUT_CH + cb0 + rc] = fmaxf(acc[mt][0][r] + bias0, 0.0f);
        out[(long long)m * OUT_CH + cb1 + rc] = fmaxf(acc[mt][1][r] + bias1, 0.0f);
      }
    }
  }
}

extern "C" void kernel_launch(void* const* d_in, const int* in_sizes, int n_in,
                              void* d_out, int out_size, void* d_ws, size_t ws_size,
                              hipStream_t stream) {
  const float*     x   = (const float*)d_in[0];
  const long long* ei  = (const long long*)d_in[1];
  const float*     W_l = (const float*)d_in[2];
  const float*     b_l = (const float*)d_in[3];
  const float*     W_r = (const float*)d_in[4];
  float*           out = (float*)d_out;

  const int N = in_sizes[0] / IN_CH;   // 50000
  const int E = in_sizes[1] / 2;       // 800000

  float* sum_agg = (float*)d_ws;                          // N*128 floats
  float* count   = sum_agg + (size_t)N * IN_CH;           // N floats

  const long long zn = (long long)N * IN_CH + N;
  zero_ws_kernel<<<4096, 256, 0, stream>>>(sum_agg, zn);

  const long long scatter_threads = (long long)E * 32;
  const int scatter_blocks = (int)((scatter_threads + 255) / 256);
  scatter_kernel<<<scatter_blocks, 256, 0, stream>>>(x, ei, sum_agg, count, E);

  const int gemm_blocks = (N + MTILE - 1) / MTILE;  // 782
  sage_gemm_kernel<<<gemm_blocks, 256, 0, stream>>>(x, sum_agg, count, W_l, b_l, W_r, out, N);
}